// SignatureNormalizerModule_40862318854702
// MI455X (gfx1250) — compile-verified
//
#include <hip/hip_runtime.h>
#include <hip/hip_bf16.h>

// Shapes fixed by the reference: values [64, 2048, 512] fp32.
constexpr int BATCH = 64;
constexpr int NOBS  = 2048;
constexpr int HIST  = 512;
#define EPS_F 1e-8f

typedef __attribute__((ext_vector_type(2))) float v2f;
typedef __attribute__((ext_vector_type(4))) float v4f;
typedef __attribute__((ext_vector_type(8))) float v8f;

// ---------------------------------------------------------------------------
// Kernel 1: per-observation batch statistics via V_WMMA_F32_16X16X4_F32,
// Welford merge with running stats, emit fused shift/scale for the stream pass.
// One wave (32 threads) per group of 16 observation columns -> 128 blocks.
//
// D = A(16x4, all ones) x B(4x16, batch values) + C  accumulates column sums.
// 64 batch rows = 16 chained WMMAs of K=4. Second accumulator gets x^2 for
// the sum of squares. Since A is all-ones, only the lane<->column mapping
// (lane % 16) matters; each column's 4 rows per chunk are split between
// lane L (rows 0,1) and lane L+16 (rows 2,3).
// ---------------------------------------------------------------------------
__global__ __launch_bounds__(32) void sig_stats_kernel(
    const float* __restrict__ values,
    const float* __restrict__ means,
    const float* __restrict__ M2s,
    const int*   __restrict__ counts,
    const int*   __restrict__ sig_ids,
    float* __restrict__ shift,
    float* __restrict__ scale)
{
    const int lane = threadIdx.x;              // 0..31, wave32
    const int n0   = blockIdx.x * 16;          // column group base
    const int col  = n0 + (lane & 15);
    const long long strideB = (long long)NOBS * HIST;      // batch stride
    const float* base = values + (long long)col * HIST + (HIST - 1);
    const int rsel = (lane >> 4) * 2;          // lanes 0-15: rows {0,1}; 16-31: {2,3}

    v2f a = {1.0f, 1.0f};                      // ones A-matrix (16x4)
    v8f cs = {};                               // sum accumulator
    v8f cq = {};                               // sum-of-squares accumulator

#pragma unroll
    for (int kk = 0; kk < 16; ++kk) {
        const int r = kk * 4 + rsel;
        const float x0 = base[(long long)r * strideB];
        const float x1 = base[(long long)(r + 1) * strideB];
        v2f b  = {x0, x1};
        v2f b2 = {x0 * x0, x1 * x1};
        // 8 args: (neg_a, A, neg_b, B, c_mod, C, reuse_a, reuse_b)
        cs = __builtin_amdgcn_wmma_f32_16x16x4_f32(false, a, false, b,
                                                   (short)0, cs, false, false);
        cq = __builtin_amdgcn_wmma_f32_16x16x4_f32(false, a, false, b2,
                                                   (short)0, cq, false, false);
    }

    // All D rows are identical (A rows identical); lane L holds column n0+L%16.
    const float S = cs[0];
    const float Q = cq[0];

    if (lane < 16) {
        const int n   = col;
        const int sid = sig_ids[n];
        const float c   = (float)counts[sid];
        const float m   = means[sid];
        const float M2  = M2s[sid];

        const float mean_new = S * (1.0f / (float)BATCH);
        // sum((v-mean)^2) == Q - S^2/BATCH; clamp tiny negative rounding.
        const float M2_new   = fmaxf(Q - S * mean_new, 0.0f);

        const float delta     = mean_new - m;
        const float total     = c + (float)BATCH;
        const float m_merged  = m + delta * ((float)BATCH / total);
        const float M2_merged = M2 + M2_new + delta * delta * c * (float)BATCH / total;
        const float var       = M2_merged / total;
        const float stdv      = sqrtf(var + EPS_F);
        const bool  ok        = (total >= 2.0f);

        // Fold the ok-mask: identity transform when !ok.
        shift[n] = ok ? m_merged : 0.0f;
        scale[n] = ok ? (1.0f / stdv) : 1.0f;
    }
}

// ---------------------------------------------------------------------------
// Kernel 2: bandwidth-bound streaming normalize, float4 per thread,
// non-temporal b128 loads/stores (512 MB traffic streams past the 192 MB L2).
// idx4 layout: h4 = idx4 & 127 (HIST/4), n = (idx4 >> 7) & 2047, b = idx4 >> 18.
// ---------------------------------------------------------------------------
__global__ __launch_bounds__(256) void sig_norm_kernel(
    const float* __restrict__ values,
    const float* __restrict__ shift,
    const float* __restrict__ scale,
    float* __restrict__ out)
{
    const int idx = blockIdx.x * 256 + threadIdx.x;   // float4 index, < 2^24
    const int n   = (idx >> 7) & (NOBS - 1);          // HIST/4 = 128 = 2^7

    const float sh = shift[n];
    const float sc = scale[n];

    const v4f* __restrict__ in4 = (const v4f*)values;
    v4f* __restrict__ out4 = (v4f*)out;

    v4f x = __builtin_nontemporal_load(&in4[idx]);
    v4f y = (x - sh) * sc;
    __builtin_nontemporal_store(y, &out4[idx]);
}

extern "C" void kernel_launch(void* const* d_in, const int* in_sizes, int n_in,
                              void* d_out, int out_size, void* d_ws, size_t ws_size,
                              hipStream_t stream)
{
    const float* values  = (const float*)d_in[0];
    const float* means   = (const float*)d_in[1];
    const float* M2s     = (const float*)d_in[2];
    const int*   counts  = (const int*)d_in[3];
    const int*   sig_ids = (const int*)d_in[4];
    float* out = (float*)d_out;

    float* shift = (float*)d_ws;        // [NOBS]
    float* scale = shift + NOBS;        // [NOBS]   (16 KB total of d_ws)

    // Stage 1: stats + merge -> shift/scale (128 waves, WMMA reduction).
    sig_stats_kernel<<<NOBS / 16, 32, 0, stream>>>(values, means, M2s, counts,
                                                   sig_ids, shift, scale);

    // Stage 2: streaming normalize (64*2048*512 / 4 float4s / 256 threads).
    const int total4 = BATCH * NOBS * (HIST / 4);
    sig_norm_kernel<<<total4 / 256, 256, 0, stream>>>(values, shift, scale, out);
}